// BoundaryLoss_15805479649595
// MI455X (gfx1250) — compile-verified
//
#include <hip/hip_runtime.h>
#include <stdint.h>

// ---------------------------------------------------------------------------
// BoundaryLoss fused kernel for gfx1250 (MI455X).
//   loss = mean( (clip(|L*softmax(pred)|,0,1) - clip(|L*onehot(t)|,0,1))^2 )
// Bandwidth-bound (159MB pred + 8MB target -> 1 scalar): single fused pass,
// async global->LDS DMA (ASYNCcnt) for the 19-channel halo tiles.
// Tile 32x16 (512 thr): halo/softmax duplication factor 612/512 = 1.195.
// ---------------------------------------------------------------------------

#define CCH     19
#define IMG     512
#define TILE_W  32
#define TILE_H  16
#define HALO_W  (TILE_W + 2)            // 34
#define HALO_H  (TILE_H + 2)            // 18
#define NPIX    (HALO_W * HALO_H)       // 612
#define NLOAD   (CCH * NPIX)            // 11628
#define THREADS 512                      // 16 waves of 32
#define NWAVES  (THREADS / 32)
#define GRID_X  (IMG / TILE_W)          // 16
#define GRID_Y  (IMG / TILE_H)          // 32
#define GRID_Z  8
#define NBLOCKS (GRID_X * GRID_Y * GRID_Z)  // 4096
#define INV_N   (1.0f / (8.0f * 19.0f * 512.0f * 512.0f))

// Async DMA one dword global -> LDS (per-lane addresses). Tracked by ASYNCcnt.
__device__ __forceinline__ void async_ld_b32(uint32_t lds_byte_off, const void* gaddr) {
    asm volatile("global_load_async_to_lds_b32 %0, %1, off"
                 :: "v"(lds_byte_off), "v"(gaddr)
                 : "memory");
}

__device__ __forceinline__ void wait_async_zero() {
#if __has_builtin(__builtin_amdgcn_s_wait_asynccnt)
    __builtin_amdgcn_s_wait_asynccnt(0);
#else
    asm volatile("s_wait_asynccnt 0" ::: "memory");
#endif
}

// Generic (flat) pointer to a __shared__ object: low 32 bits are the LDS offset.
__device__ __forceinline__ uint32_t lds_off(const void* p) {
    return (uint32_t)(uintptr_t)p;
}

__global__ __launch_bounds__(THREADS)
void boundary_loss_tile(const float* __restrict__ pred,
                        const int*   __restrict__ target,
                        float*       __restrict__ partials,
                        float*       __restrict__ out,
                        int use_atomic)
{
    __shared__ float sp[CCH][HALO_H][HALO_W];   // softmax tile (in-place), 46512 B
    __shared__ int   st[HALO_H][HALO_W];        // target tile, -1 = outside image
    __shared__ float wsum[NWAVES];

    const int tid = threadIdx.x;
    const int bx = blockIdx.x, by = blockIdx.y, b = blockIdx.z;
    const int x0 = bx * TILE_W - 1;
    const int y0 = by * TILE_H - 1;

    // ---- Phase 1: async DMA pred halo tile (19 ch) + target tile into LDS ----
    for (int i = tid; i < NLOAD; i += THREADS) {
        int ch = i / NPIX;
        int r  = i - ch * NPIX;
        int ry = r / HALO_W;
        int rx = r - ry * HALO_W;
        int gx = x0 + rx, gy = y0 + ry;
        if ((unsigned)gx < IMG && (unsigned)gy < IMG) {
            const float* g = pred + ((((size_t)b * CCH + ch) * IMG + gy) * IMG + gx);
            async_ld_b32(lds_off(&sp[ch][ry][rx]), g);
        } else {
            sp[ch][ry][rx] = 0.0f;   // SAME zero padding of the conv input
        }
    }
    for (int i = tid; i < NPIX; i += THREADS) {
        int ry = i / HALO_W;
        int rx = i - ry * HALO_W;
        int gx = x0 + rx, gy = y0 + ry;
        if ((unsigned)gx < IMG && (unsigned)gy < IMG) {
            const int* g = target + (((size_t)b * IMG + gy) * IMG + gx);
            async_ld_b32(lds_off(&st[ry][rx]), g);
        } else {
            st[ry][rx] = -1;         // matches no class -> onehot contributes 0
        }
    }
    wait_async_zero();
    __syncthreads();

    // ---- Phase 2: in-place softmax over channels at every in-image halo pixel ----
    for (int i = tid; i < NPIX; i += THREADS) {
        int ry = i / HALO_W;
        int rx = i - ry * HALO_W;
        int gx = x0 + rx, gy = y0 + ry;
        if ((unsigned)gx < IMG && (unsigned)gy < IMG) {
            float m = sp[0][ry][rx];
            #pragma unroll
            for (int c = 1; c < CCH; ++c) m = fmaxf(m, sp[c][ry][rx]);
            float s = 0.0f;
            #pragma unroll
            for (int c = 0; c < CCH; ++c) {
                float e = __expf(sp[c][ry][rx] - m);
                sp[c][ry][rx] = e;
                s += e;
            }
            float inv = 1.0f / s;
            #pragma unroll
            for (int c = 0; c < CCH; ++c) sp[c][ry][rx] *= inv;
        }
    }
    __syncthreads();

    // ---- Phase 3: Laplacian boundary of p and of onehot(t); squared diff ----
    const int rx = tid & (TILE_W - 1);
    const int ry = tid >> 5;                  // [0, TILE_H)
    const int hx = rx + 1, hy = ry + 1;

    int t9[9];
    #pragma unroll
    for (int dy = 0; dy < 3; ++dy)
        #pragma unroll
        for (int dx = 0; dx < 3; ++dx)
            t9[dy * 3 + dx] = st[hy + dy - 1][hx + dx - 1];
    const int ctr = t9[4];

    float acc = 0.0f;
    #pragma unroll
    for (int ch = 0; ch < CCH; ++ch) {
        float s9 = 0.0f;
        #pragma unroll
        for (int dy = 0; dy < 3; ++dy)
            #pragma unroll
            for (int dx = 0; dx < 3; ++dx)
                s9 += sp[ch][hy + dy - 1][hx + dx - 1];
        float cv = sp[ch][hy][hx];
        float yp = fmaf(9.0f, cv, -s9);       // == 8*center - sum(8 neighbors)
        float pb = fminf(fabsf(yp), 1.0f);

        int cnt = 0;
        #pragma unroll
        for (int k = 0; k < 9; ++k) cnt += (t9[k] == ch) ? 1 : 0;
        // onehot boundary: center==ch -> clip(|9-cnt|); else clip(cnt)
        float tb = (ctr == ch) ? ((cnt < 9) ? 1.0f : 0.0f)
                               : ((cnt > 0) ? 1.0f : 0.0f);
        float d = pb - tb;
        acc = fmaf(d, d, acc);
    }

    // ---- Phase 4: block reduction (wave32 shuffles + wave partials) ----
    #pragma unroll
    for (int o = 16; o > 0; o >>= 1) acc += __shfl_down(acc, o, 32);
    if ((tid & 31) == 0) wsum[tid >> 5] = acc;
    __syncthreads();
    if (tid == 0) {
        float s = 0.0f;
        #pragma unroll
        for (int w = 0; w < NWAVES; ++w) s += wsum[w];
        if (use_atomic) {
            atomicAdd(out, s * INV_N);
        } else {
            int bid = (b * (int)gridDim.y + by) * (int)gridDim.x + bx;
            partials[bid] = s;
        }
    }
}

__global__ __launch_bounds__(256)
void reduce_partials(const float* __restrict__ partials, float* __restrict__ out, int n)
{
    __shared__ float wsum[8];
    float a = 0.0f;
    for (int i = threadIdx.x; i < n; i += 256) a += partials[i];
    #pragma unroll
    for (int o = 16; o > 0; o >>= 1) a += __shfl_down(a, o, 32);
    if ((threadIdx.x & 31) == 0) wsum[threadIdx.x >> 5] = a;
    __syncthreads();
    if (threadIdx.x == 0) {
        float s = 0.0f;
        #pragma unroll
        for (int w = 0; w < 8; ++w) s += wsum[w];
        out[0] = s * INV_N;
    }
}

__global__ void zero_out(float* out) { out[0] = 0.0f; }

extern "C" void kernel_launch(void* const* d_in, const int* in_sizes, int n_in,
                              void* d_out, int out_size, void* d_ws, size_t ws_size,
                              hipStream_t stream)
{
    const float* pred   = (const float*)d_in[0];
    const int*   target = (const int*)d_in[1];
    float*       out    = (float*)d_out;

    dim3 grid(GRID_X, GRID_Y, GRID_Z);
    const bool use_ws = (ws_size >= (size_t)NBLOCKS * sizeof(float));

    if (use_ws) {
        float* partials = (float*)d_ws;
        boundary_loss_tile<<<grid, THREADS, 0, stream>>>(pred, target, partials, out, 0);
        reduce_partials<<<1, 256, 0, stream>>>(partials, out, NBLOCKS);
    } else {
        zero_out<<<1, 1, 0, stream>>>(out);
        boundary_loss_tile<<<grid, THREADS, 0, stream>>>(pred, target, nullptr, out, 1);
    }
}